// PtsManipulator_59768764891317
// MI455X (gfx1250) — compile-verified
//
#include <hip/hip_runtime.h>

#define HW    9216      // 96*96 pixels == number of points
#define NC    64
#define SZ    96
#define PPP   8
#define NTILE 576       // HW/16
#define R2    0.0009765625f   // (1.5/96*2)^2

typedef __attribute__((ext_vector_type(2))) float v2f;
typedef __attribute__((ext_vector_type(8))) float v8f;
typedef unsigned long long u64;
typedef int vi4 __attribute__((ext_vector_type(4)));

#if __has_builtin(__builtin_amdgcn_global_load_async_to_lds_b128) && \
    __has_builtin(__builtin_amdgcn_s_wait_asynccnt)
#define USE_ASYNC_LDS 1
// pointer-to-AS1-int4 / pointer-to-AS3-int4, per the builtin's signature
typedef __attribute__((address_space(1))) vi4* gvi4_ptr;
typedef __attribute__((address_space(3))) vi4* lvi4_ptr;
#endif

// ---------------------------------------------------------------------------
// Phase 0: project all points, emit packed WMMA A-operand rows and z.
//   ptA[p] = { -2x, -2y, x^2+y^2, 1 }   (K=4 row for the rank-4 distance GEMM)
//   zarr[p] = z
// ---------------------------------------------------------------------------
__global__ void pts_project_kernel(const float* __restrict__ pred,
                                   const float* __restrict__ Kinv,
                                   const float* __restrict__ RT2,
                                   const float* __restrict__ RTi1,
                                   const float* __restrict__ Kmat,
                                   float* __restrict__ ptA,
                                   float* __restrict__ zarr) {
    const int b = blockIdx.y;
    const int p = blockIdx.x * blockDim.x + threadIdx.x;
    if (p >= HW) return;
    const int row = p / SZ, col = p - row * SZ;
    const float xs = (float)col / 95.0f * 2.0f - 1.0f;
    const float ys = (float)row / 95.0f * 2.0f - 1.0f;
    const float d  = pred[b * HW + p];
    float pc[4] = { xs * d, -ys * d, -d, 1.0f };

    const float* Ki = Kinv + b * 16;
    const float* Ra = RT2  + b * 16;
    const float* Rb = RTi1 + b * 16;
    const float* Km = Kmat + b * 16;

    float cam[4];
#pragma unroll
    for (int i = 0; i < 4; ++i)
        cam[i] = Ki[i*4+0]*pc[0] + Ki[i*4+1]*pc[1] + Ki[i*4+2]*pc[2] + Ki[i*4+3]*pc[3];

    float RT[16];
#pragma unroll
    for (int i = 0; i < 4; ++i)
#pragma unroll
        for (int k = 0; k < 4; ++k)
            RT[i*4+k] = Ra[i*4+0]*Rb[0*4+k] + Ra[i*4+1]*Rb[1*4+k]
                      + Ra[i*4+2]*Rb[2*4+k] + Ra[i*4+3]*Rb[3*4+k];

    float wr[4];
#pragma unroll
    for (int i = 0; i < 4; ++i)
        wr[i] = RT[i*4+0]*cam[0] + RT[i*4+1]*cam[1] + RT[i*4+2]*cam[2] + RT[i*4+3]*cam[3];

    float xyp0 = Km[0]*wr[0] + Km[1]*wr[1] + Km[2]*wr[2]  + Km[3]*wr[3];
    float xyp1 = Km[4]*wr[0] + Km[5]*wr[1] + Km[6]*wr[2]  + Km[7]*wr[3];
    float xyp2 = Km[8]*wr[0] + Km[9]*wr[1] + Km[10]*wr[2] + Km[11]*wr[3];

    const bool mask = fabsf(xyp2) < 0.01f;
    const float zs  = mask ? 0.01f : xyp2;
    float x =  xyp0 / zs;
    float y = -xyp1 / zs;
    float z = -xyp2;
    if (mask) { x = 10.0f; y = -10.0f; z = 10.0f; }

    float4 A = make_float4(-2.0f * x, -2.0f * y, x * x + y * y, 1.0f);
    reinterpret_cast<float4*>(ptA)[b * HW + p] = A;
    zarr[b * HW + p] = z;
}

// ---------------------------------------------------------------------------
// Phase 1: per-pixel top-8-by-z splat + compositing + 64-channel gather.
// One wave owns a 16-pixel tile (WMMA N dim). 8 waves / block, independent.
// Selection keys: u64 = (z_bits << 32) | point_index  (z > 0 => bit-ordered),
// giving stable lax.top_k tie-breaking in a single branchless u64 compare.
// ---------------------------------------------------------------------------
__global__ void __launch_bounds__(256)
pts_splat_kernel(const float* __restrict__ ptA,
                 const float* __restrict__ zarr,
                 const float* __restrict__ alphas,
                 const float* __restrict__ src,
                 float* __restrict__ out) {
    __shared__ float lds_z[HW];            // 36 KB of the 320 KB WGP LDS

    const int b    = blockIdx.y;
    const int wave = threadIdx.x >> 5;
    const int lane = threadIdx.x & 31;
    const int tile = blockIdx.x * 8 + wave;        // 0..575
    const int n    = lane & 15;                    // pixel column within tile
    const int hi   = lane >> 4;                    // K-half / row-half select
    const int pix  = tile * 16 + n;                // global pixel id
    const int prow = pix / SZ, pcol = pix - prow * SZ;

    const float* Ab = ptA  + (size_t)b * HW * 4;
    const float* Zb = zarr + (size_t)b * HW;

    // ---- stage z[] into LDS (async DMA path; ASYNCcnt-tracked) ----
#ifdef USE_ASYNC_LDS
#pragma unroll
    for (int i = 0; i < HW / 4 / 256; ++i) {
        const int e = (i * 256 + threadIdx.x) * 4;   // float4 chunk
        __builtin_amdgcn_global_load_async_to_lds_b128(
            (gvi4_ptr)(Zb + e), (lvi4_ptr)(lds_z + e), 0, 0);
    }
    __builtin_amdgcn_s_wait_asynccnt(0);
    __syncthreads();
#else
#pragma unroll
    for (int i = 0; i < HW / 4 / 256; ++i) {
        const int e = i * 256 + threadIdx.x;
        reinterpret_cast<float4*>(lds_z)[e] =
            reinterpret_cast<const float4*>(Zb)[e];
    }
    __syncthreads();
#endif

    const float px = 1.0f - (2.0f * (float)pcol + 1.0f) / (float)SZ;
    const float py = 1.0f - (2.0f * (float)prow + 1.0f) / (float)SZ;
    const float px2py2 = px * px + py * py;

    // B operand column for this pixel: [px, py, 1, px^2+py^2], split K-halves.
    v2f bop;
    if (hi == 0) { bop.x = px;   bop.y = py; }
    else         { bop.x = 1.0f; bop.y = px2py2; }

    u64 bk[PPP];   // sorted ascending (z_bits, idx); empty = inf key
    const u64 KINF = ((u64)0x7F800000u << 32);
#pragma unroll
    for (int k = 0; k < PPP; ++k) bk[k] = KINF;

    // 2-stage software pipeline on the A-operand load (stride 256 B / tile).
    const float* ap = Ab + (size_t)n * 4 + hi * 2;
    v2f a_cur = *reinterpret_cast<const v2f*>(ap);

    for (int t = 0; t < NTILE; ++t) {
        const int tn = (t + 1 < NTILE) ? (t + 1) : (NTILE - 1);
        v2f a_nxt = *reinterpret_cast<const v2f*>(ap + (size_t)tn * 64);

        const int tp = (t + 8 < NTILE) ? (t + 8) : (NTILE - 1);
        __builtin_prefetch(ap + (size_t)tp * 64, 0, 1);

        v8f c = {0.0f, 0.0f, 0.0f, 0.0f, 0.0f, 0.0f, 0.0f, 0.0f};
        // d2[point m, pixel n] tile via rank-4 f32 WMMA
        c = __builtin_amdgcn_wmma_f32_16x16x4_f32(false, a_cur, false, bop,
                                                  (short)0, c, false, false);

        // fast gate: any of this lane's 8 candidates inside the radius?
        const float dmin = fminf(fminf(fminf(c[0], c[1]), fminf(c[2], c[3])),
                                 fminf(fminf(c[4], c[5]), fminf(c[6], c[7])));
        if (dmin < R2) {
            const int mb = t * 16;
            const float* zp = lds_z + mb + hi * 8;
            float4 z0 = *reinterpret_cast<const float4*>(zp);      // ds_load_b128
            float4 z1 = *reinterpret_cast<const float4*>(zp + 4);
            float zr[8] = { z0.x, z0.y, z0.z, z0.w, z1.x, z1.y, z1.z, z1.w };

#pragma unroll
            for (int r = 0; r < 8; ++r) {
                // Poison invalid candidates to UINT64_MAX via a 32-bit cndmask
                // mask OR'd into both key halves (no exec-mask branches).
                const bool invalid = !((c[r] < R2) && (zr[r] > 0.0f));
                const unsigned m = invalid ? 0xFFFFFFFFu : 0u;   // v_cndmask_b32
                u64 k2 = (((u64)(__float_as_uint(zr[r]) | m) << 32)
                          | (unsigned)((mb + hi * 8 + r) | (int)m));
#pragma unroll
                for (int s = 0; s < PPP; ++s) {     // branchless sorted insert
                    const u64 lo = k2 < bk[s] ? k2 : bk[s];   // v_min_u64
                    const u64 hh = k2 < bk[s] ? bk[s] : k2;   // v_max_u64
                    bk[s] = lo; k2 = hh;
                }
            }
        }
        a_cur = a_nxt;
    }

    // Merge with partner lane (lane ^ 16): both halves of this pixel's points.
#pragma unroll
    for (int k = 0; k < PPP; ++k) {
        u64 k2 = __shfl_xor(bk[k], 16, 32);
#pragma unroll
        for (int s = 0; s < PPP; ++s) {
            const u64 lo = k2 < bk[s] ? k2 : bk[s];
            const u64 hh = k2 < bk[s] ? bk[s] : k2;
            bk[s] = lo; k2 = hh;
        }
    }

    // Compositing weights: w = (1 - sqrt(clip(d2/r2, 0.001, 1))) * alpha * ok
    const float* alp = alphas + (size_t)b * HW;
    float wt[PPP];
    int   idx[PPP];
    float trans = 1.0f;
#pragma unroll
    for (int k = 0; k < PPP; ++k) {
        const bool ok = (unsigned)(bk[k] >> 32) < 0x7F800000u;
        const int  id = ok ? (int)(unsigned)bk[k] : 0;
        idx[k] = id;
        const float4 A = reinterpret_cast<const float4*>(Ab)[id];
        const float d2 = fmaf(A.x, px, fmaf(A.y, py, A.z + px2py2));
        float dist = d2 * (1.0f / R2);
        dist = fminf(fmaxf(dist, 0.001f), 1.0f);
        float w = 1.0f - sqrtf(dist);
        w *= ok ? alp[id] : 0.0f;
        wt[k] = w * trans;
        trans *= (1.0f - w);
    }

    // Channel gather: lane handles (pixel = n, channels it*2+hi) -> all 64 ch.
    const float* sb = src + (size_t)b * NC * HW;
    float*       ob = out + (size_t)b * NC * HW;
    for (int it = 0; it < 32; ++it) {
        const int ch = it * 2 + hi;
        const float* sc = sb + (size_t)ch * HW;
        float acc = 0.0f;
#pragma unroll
        for (int k = 0; k < PPP; ++k)
            acc += sc[idx[k]] * wt[k];
        ob[(size_t)ch * HW + pix] = acc;
    }
}

// ---------------------------------------------------------------------------
extern "C" void kernel_launch(void* const* d_in, const int* in_sizes, int n_in,
                              void* d_out, int out_size, void* d_ws, size_t ws_size,
                              hipStream_t stream) {
    const float* alphas = (const float*)d_in[0];
    const float* src    = (const float*)d_in[1];
    const float* pred   = (const float*)d_in[2];
    const float* Kmat   = (const float*)d_in[3];
    const float* Kinv   = (const float*)d_in[4];
    // d_in[5] = RT_cam1 (unused by reference math)
    const float* RTi1   = (const float*)d_in[6];
    const float* RT2    = (const float*)d_in[7];
    // d_in[8] = RTinv_cam2 (unused)

    float* ws   = (float*)d_ws;
    float* ptA  = ws;                    // 2 * 9216 * 4 floats
    float* zarr = ws + 2 * HW * 4;       // 2 * 9216 floats

    pts_project_kernel<<<dim3(HW / 256, 2), 256, 0, stream>>>(
        pred, Kinv, RT2, RTi1, Kmat, ptA, zarr);

    pts_splat_kernel<<<dim3(NTILE / 8, 2), 256, 0, stream>>>(
        ptA, zarr, alphas, src, (float*)d_out);
}